// MessagePassingLayer_88364657148618
// MI455X (gfx1250) — compile-verified
//
#include <hip/hip_runtime.h>

// ---------------------------------------------------------------------------
// MessagePassing layer for MI455X (gfx1250):
//   proj = relu(h @ W^T)   fp32 WMMA (16x16x4), async-to-LDS staging,
//                          64-row blocks (4 tiles/wave) to amortize W staging.
//   out  = segment_sum(proj[src], dst)
//          CSR-by-dst (histogram + scan + reorder) -> atomic-free gather-sum.
// ---------------------------------------------------------------------------

typedef float v2f __attribute__((ext_vector_type(2)));
typedef float v8f __attribute__((ext_vector_type(8)));

#define IN_DIM   128
#define OUT_DIM  128
#define LDS_STR  132   // 132 mod 64 banks = 4 -> conflict-free B64 fragment loads
#define ROWS_PB  64    // rows per block (4 row-tiles per wave)

// Async copy of 16 bytes global -> LDS (CDNA5 TDM-lite path, ASYNCcnt-tracked).
__device__ __forceinline__ void async_g2l_b128(void* lds_ptr, const void* gptr) {
    const unsigned lds_addr = (unsigned)(size_t)lds_ptr;   // low 32 bits = LDS offset
    asm volatile("global_load_async_to_lds_b128 %0, %1, off"
                 :: "v"(lds_addr), "v"(gptr) : "memory");
}
__device__ __forceinline__ void wait_asynccnt0() {
    asm volatile("s_wait_asynccnt 0x0" ::: "memory");
}

// ---------------------------------------------------------------------------
// GEMM kernel: proj = relu(h @ W^T) using fp32 WMMA (16x16x4).
// Block = 64 rows x 128 cols; 8 waves; wave owns col tile, 4 row tiles.
// ---------------------------------------------------------------------------
__global__ __launch_bounds__(256) void mp_gemm_relu_wmma(
    const float* __restrict__ h,   // [N, 128]
    const float* __restrict__ W,   // [128, 128] (out, in)
    float* __restrict__ proj,      // [N, 128]
    int n_nodes)
{
    __shared__ float Ws[OUT_DIM * LDS_STR];    // W[n][k] at Ws[n*LDS_STR + k]
    __shared__ float hs[ROWS_PB * LDS_STR];    // h[rowBase+m][k]

    const int tid     = threadIdx.x;
    const int wave    = tid >> 5;
    const int lane    = tid & 31;
    const int rowBase = blockIdx.x * ROWS_PB;

    // Stage W (16384 floats): async b128, 16 per thread.
    for (int i = tid; i < OUT_DIM * (IN_DIM / 4); i += 256) {
        const int r  = i >> 5;
        const int c4 = i & 31;
        async_g2l_b128(&Ws[r * LDS_STR + c4 * 4], W + r * IN_DIM + c4 * 4);
    }
    // Stage 64 h rows (8192 floats): async b128; OOB rows zero-filled via DS.
    for (int i = tid; i < ROWS_PB * (IN_DIM / 4); i += 256) {
        const int r  = i >> 5;
        const int c4 = i & 31;
        const int gr = rowBase + r;
        float* lp = &hs[r * LDS_STR + c4 * 4];
        if (gr < n_nodes) {
            async_g2l_b128(lp, h + (size_t)gr * IN_DIM + c4 * 4);
        } else {
            lp[0] = 0.f; lp[1] = 0.f; lp[2] = 0.f; lp[3] = 0.f;
        }
    }
    wait_asynccnt0();
    __syncthreads();

    // Fragment coords (CDNA5 f32 16x4 A layout):
    //   lanes 0-15: K = k0+{0,1}; lanes 16-31: K = k0+{2,3}; M/N = lane&15.
    const int colBase = wave * 16;
    const int mrow    = lane & 15;
    const int khalf   = (lane >> 4) * 2;

    v8f acc[4] = {v8f{}, v8f{}, v8f{}, v8f{}};
    #pragma unroll
    for (int k0 = 0; k0 < IN_DIM; k0 += 4) {
        const v2f b  = *(const v2f*)&Ws[(colBase + mrow) * LDS_STR + k0 + khalf];
        #pragma unroll
        for (int t = 0; t < 4; ++t) {
            const v2f a = *(const v2f*)&hs[(mrow + 16 * t) * LDS_STR + k0 + khalf];
            acc[t] = __builtin_amdgcn_wmma_f32_16x16x4_f32(
                false, a, false, b, (short)0, acc[t], false, false);
        }
    }

    // C/D layout: VGPR v -> M = v + 8*(lane>=16), N = lane&15. ReLU + store.
    const int nIdx = colBase + mrow;
    const int rOff = (lane >> 4) * 8;
    #pragma unroll
    for (int t = 0; t < 4; ++t) {
        #pragma unroll
        for (int v = 0; v < 8; ++v) {
            const int gr = rowBase + 16 * t + rOff + v;
            if (gr < n_nodes) {
                float x = acc[t][v];
                proj[(size_t)gr * OUT_DIM + nIdx] = x > 0.f ? x : 0.f;
            }
        }
    }
}

// ---------------------------------------------------------------------------
// CSR build: zero counts -> histogram -> 3-kernel exclusive scan -> reorder.
// ---------------------------------------------------------------------------
__global__ __launch_bounds__(256) void mp_zero_ints(int* __restrict__ p, int n) {
    int i = blockIdx.x * 256 + threadIdx.x;
    if (i < n) p[i] = 0;
}

__global__ __launch_bounds__(256) void mp_hist(const int* __restrict__ dst,
                                               int* __restrict__ cnt, int n_edges) {
    int e = blockIdx.x * 256 + threadIdx.x;
    if (e < n_edges) atomicAdd(&cnt[dst[e]], 1);
}

__global__ __launch_bounds__(256) void mp_scan1(const int* __restrict__ cnt,
                                                int* __restrict__ offs,
                                                int* __restrict__ bsum, int n) {
    __shared__ int s[256];
    const int tid  = threadIdx.x;
    const int base = blockIdx.x * 1024 + tid * 4;

    int v[4]; int sum = 0;
    #pragma unroll
    for (int j = 0; j < 4; ++j) {
        const int idx = base + j;
        v[j] = (idx < n) ? cnt[idx] : 0;
        sum += v[j];
    }
    s[tid] = sum;
    __syncthreads();
    for (int off = 1; off < 256; off <<= 1) {
        int t = 0;
        if (tid >= off) t = s[tid - off];
        __syncthreads();
        s[tid] += t;
        __syncthreads();
    }
    int run = s[tid] - sum;
    #pragma unroll
    for (int j = 0; j < 4; ++j) {
        const int idx = base + j;
        if (idx < n) offs[idx] = run;
        run += v[j];
    }
    if (tid == 255) bsum[blockIdx.x] = s[255];
}

__global__ __launch_bounds__(256) void mp_scan2(int* __restrict__ bsum, int nb) {
    __shared__ int s[256];
    const int tid = threadIdx.x;
    const int v = (tid < nb) ? bsum[tid] : 0;
    s[tid] = v;
    __syncthreads();
    for (int off = 1; off < 256; off <<= 1) {
        int t = 0;
        if (tid >= off) t = s[tid - off];
        __syncthreads();
        s[tid] += t;
        __syncthreads();
    }
    if (tid < nb) bsum[tid] = s[tid] - v;
}

__global__ __launch_bounds__(256) void mp_scan3(int* __restrict__ offs,
                                                const int* __restrict__ bsum,
                                                int* __restrict__ cursor,
                                                int n, int n_edges) {
    int i = blockIdx.x * 256 + threadIdx.x;
    if (i < n) {
        const int o = offs[i] + bsum[i >> 10];
        offs[i]   = o;
        cursor[i] = o;
    }
    if (i == 0) offs[n] = n_edges;
}

__global__ __launch_bounds__(256) void mp_reorder(const int* __restrict__ src,
                                                  const int* __restrict__ dst,
                                                  int* __restrict__ cursor,
                                                  int* __restrict__ esrc, int n_edges) {
    int e = blockIdx.x * 256 + threadIdx.x;
    if (e < n_edges) {
        const int p = atomicAdd(&cursor[dst[e]], 1);
        esrc[p] = src[e];
    }
}

// ---------------------------------------------------------------------------
// Atomic-free gather-sum: one wave per node, lane owns 4 columns (float4).
// ---------------------------------------------------------------------------
__global__ __launch_bounds__(256) void mp_node_gather(
    const float* __restrict__ proj,
    const int*   __restrict__ offs,
    const int*   __restrict__ esrc,
    float*       __restrict__ out,
    int n_nodes)
{
    const int node = blockIdx.x * 8 + (threadIdx.x >> 5);
    if (node >= n_nodes) return;
    const int lane = threadIdx.x & 31;

    const int j0 = offs[node];
    const int j1 = offs[node + 1];

    float4 acc; acc.x = 0.f; acc.y = 0.f; acc.z = 0.f; acc.w = 0.f;
    for (int j = j0; j < j1; ++j) {
        const int s = esrc[j];
        if (j + 1 < j1) {
            __builtin_prefetch(&proj[(size_t)esrc[j + 1] * OUT_DIM], 0, 0);
        }
        const float4 v = ((const float4*)(proj + (size_t)s * OUT_DIM))[lane];
        acc.x += v.x; acc.y += v.y; acc.z += v.z; acc.w += v.w;
    }
    ((float4*)(out + (size_t)node * OUT_DIM))[lane] = acc;
}

// ---------------------------------------------------------------------------
// Fallback path (small workspace): zero output + per-edge atomic scatter.
// ---------------------------------------------------------------------------
__global__ __launch_bounds__(256) void mp_zero_f4(float4* __restrict__ out, int n4) {
    int i = blockIdx.x * 256 + threadIdx.x;
    if (i < n4) {
        float4 z; z.x = 0.f; z.y = 0.f; z.z = 0.f; z.w = 0.f;
        out[i] = z;
    }
}

__global__ __launch_bounds__(256) void mp_edge_scatter(
    const float* __restrict__ proj,
    const int*   __restrict__ src,
    const int*   __restrict__ dst,
    float*       __restrict__ out,
    int n_edges)
{
    const int e = blockIdx.x * 8 + (threadIdx.x >> 5);
    if (e >= n_edges) return;
    const int lane = threadIdx.x & 31;
    const int s = src[e];
    const int d = dst[e];
    const float4 v = ((const float4*)(proj + (size_t)s * OUT_DIM))[lane];
    float* op = out + (size_t)d * OUT_DIM + lane * 4;
    atomicAdd(op + 0, v.x);
    atomicAdd(op + 1, v.y);
    atomicAdd(op + 2, v.z);
    atomicAdd(op + 3, v.w);
}

// ---------------------------------------------------------------------------
extern "C" void kernel_launch(void* const* d_in, const int* in_sizes, int n_in,
                              void* d_out, int out_size, void* d_ws, size_t ws_size,
                              hipStream_t stream) {
    const float* h   = (const float*)d_in[0];
    const float* W   = (const float*)d_in[1];
    const int*   src = (const int*)d_in[2];
    const int*   dst = (const int*)d_in[3];
    float* out = (float*)d_out;

    const int n_nodes = in_sizes[0] / IN_DIM;    // 100000
    const int n_edges = in_sizes[2];             // 800000
    const int nb      = (n_nodes + 1023) / 1024; // scan blocks (<=256 required)

    char* wsp = (char*)d_ws;
    auto align256 = [](size_t x) { return (x + 255) & ~(size_t)255; };
    const size_t projB = align256((size_t)n_nodes * OUT_DIM * sizeof(float));
    const size_t cntB  = align256((size_t)n_nodes * sizeof(int));
    const size_t offB  = align256((size_t)(n_nodes + 1) * sizeof(int));
    const size_t curB  = align256((size_t)n_nodes * sizeof(int));
    const size_t bsumB = align256((size_t)nb * sizeof(int));
    const size_t esrcB = align256((size_t)n_edges * sizeof(int));
    const size_t need  = projB + cntB + offB + curB + bsumB + esrcB;

    float* proj  = (float*)wsp;
    int* cnt     = (int*)(wsp + projB);
    int* offs    = (int*)(wsp + projB + cntB);
    int* cursor  = (int*)(wsp + projB + cntB + offB);
    int* bsum    = (int*)(wsp + projB + cntB + offB + curB);
    int* esrc    = (int*)(wsp + projB + cntB + offB + curB + bsumB);

    // proj = relu(h @ W^T)
    const int rowTiles = (n_nodes + ROWS_PB - 1) / ROWS_PB;
    mp_gemm_relu_wmma<<<rowTiles, 256, 0, stream>>>(h, W, proj, n_nodes);

    const bool csr_ok = (ws_size >= need) && (nb <= 256);
    if (csr_ok) {
        mp_zero_ints<<<(n_nodes + 255) / 256, 256, 0, stream>>>(cnt, n_nodes);
        mp_hist    <<<(n_edges + 255) / 256, 256, 0, stream>>>(dst, cnt, n_edges);
        mp_scan1   <<<nb, 256, 0, stream>>>(cnt, offs, bsum, n_nodes);
        mp_scan2   <<<1, 256, 0, stream>>>(bsum, nb);
        mp_scan3   <<<(n_nodes + 255) / 256, 256, 0, stream>>>(offs, bsum, cursor,
                                                               n_nodes, n_edges);
        mp_reorder <<<(n_edges + 255) / 256, 256, 0, stream>>>(src, dst, cursor,
                                                               esrc, n_edges);
        mp_node_gather<<<(n_nodes + 7) / 8, 256, 0, stream>>>(proj, offs, esrc,
                                                              out, n_nodes);
    } else {
        const int n4 = n_nodes * OUT_DIM / 4;
        mp_zero_f4<<<(n4 + 255) / 256, 256, 0, stream>>>((float4*)out, n4);
        mp_edge_scatter<<<(n_edges + 7) / 8, 256, 0, stream>>>(proj, src, dst,
                                                               out, n_edges);
    }
}